// VisitPhenotypeLayer_25348896981008
// MI455X (gfx1250) — compile-verified
//
#include <hip/hip_runtime.h>
#include <math.h>
#include <stdint.h>

typedef float v2f __attribute__((ext_vector_type(2)));
typedef float v8f __attribute__((ext_vector_type(8)));

#define BT 8192          // B*T
#define DDIM 512
#define PNUM 2048
#define VNUM 40000
#define DFF 2148
#define DFF_PAD 2176
#define KC 50
#define KP 10
#define PC_NT 128        // d_pc column tile
#define PC_BLOCKS 128    // 2048 / 16 rows per block

// ---- d_out layout (floats), reference tuple order ----
#define OUT_OUT2   0u          // 8192*512   = 4194304
#define OUT_PROBS  4194304u    // 8192*2048  = 16777216
#define OUT_CDIST  20971520u   // scalar
#define OUT_ENT    20971521u   // 8192
#define OUT_INV    20979713u   // scalar
#define OUT_MET    20979714u   // scalar
#define OUT_VAR    20979715u   // scalar

// ---- workspace layout (floats) ----
#define WS_CTX 0u              // 4194304 (ctx, then out1 in place)
#define WS_H   4194304u        // 8192*2176 = 17825792 (padded hidden)
#define WS_RC  22020096u       // 40000
#define WS_RP  22060096u       // 2048
#define WS_PC  22062144u       // 128 per-block partial sums (d_pc)
#define WS_PPS 22064192u       // 2048 (sum near)
#define WS_PPE 22066240u       // 2048 (sum exp(-near))
#define WS_PPQ 22068288u       // 2048 (sum near^2)

__device__ __forceinline__ v8f wmma4(v2f a, v2f b, v8f c) {
  // D = A(16x4,f32) * B(4x16,f32) + C(16x16,f32)
  return __builtin_amdgcn_wmma_f32_16x16x4_f32(false, a, false, b, (short)0, c,
                                               false, false);
}

// Async memory->LDS copy, 4B per lane, GVS mode (SGPR base + VGPR byte offset).
// Tracked by ASYNCcnt; wait with s_wait_asynccnt before the barrier.
__device__ __forceinline__ void async_load_f32(uint32_t lds_byte_addr,
                                               const float* base,
                                               uint32_t byte_off) {
  asm volatile("global_load_async_to_lds_b32 %0, %1, %2"
               :
               : "v"(lds_byte_addr), "v"(byte_off), "s"(base)
               : "memory");
}
__device__ __forceinline__ void async_wait0() {
  asm volatile("s_wait_asynccnt 0x0" ::: "memory");
}

__device__ __forceinline__ float block_reduce_sum(float v, float* red, int tid) {
  red[tid] = v; __syncthreads();
  for (int s = 128; s > 0; s >>= 1) {
    if (tid < s) red[tid] += red[tid + s];
    __syncthreads();
  }
  float r = red[0]; __syncthreads();
  return r;
}

// ---------------- WMMA f32 GEMM ----------------
// MODE 0: C = (A * B^T) * conv_mask[row]   (logits -> masked logits)
// MODE 1: C = A * B                         (ctx)
// MODE 2: C = relu(A*B + bias), N padded    (FFN1 -> h)
// MODE 3: C = A*B + bias + resid, K padded  (FFN2 -> pre-LN2)
template <int MODE>
__launch_bounds__(256)
__global__ void gemm_kernel(const float* __restrict__ A, const float* __restrict__ Bm,
                            float* __restrict__ C,
                            const int* __restrict__ mask,
                            const float* __restrict__ bias,
                            const float* __restrict__ resid,
                            int K, int Kreal, int Ntrue,
                            int lda, int ldb, int ldc) {
  __shared__ float As[64][33];
  __shared__ float Bs[32][65];
  const int tid = threadIdx.x;
  const int lane = tid & 31;
  const int wave = tid >> 5;
  const int lh = lane >> 4;       // lane half: selects K pair {2lh, 2lh+1}
  const int l15 = lane & 15;
  const int m0 = (wave & 3) * 16; // 4 waves along M
  const int n0 = (wave >> 2) * 32;// 2 waves along N, each does 2 tiles
  const int rowBase = blockIdx.x * 64;
  const int nBase = blockIdx.y * 64;

  v8f acc0 = {0.f, 0.f, 0.f, 0.f, 0.f, 0.f, 0.f, 0.f};
  v8f acc1 = {0.f, 0.f, 0.f, 0.f, 0.f, 0.f, 0.f, 0.f};

  for (int kb = 0; kb < K; kb += 32) {
    // stage A tile [64 x 32] via async memory->LDS (no VGPR round trip)
    #pragma unroll
    for (int t = tid; t < 64 * 32; t += 256) {
      int r = t >> 5, c = t & 31;
      async_load_f32((uint32_t)(uintptr_t)&As[r][c], A,
                     (uint32_t)(((size_t)(rowBase + r) * lda + kb + c) * 4));
    }
    // stage B tile [32 x 64]
    if (MODE == 0) {
      #pragma unroll
      for (int t = tid; t < 64 * 32; t += 256) {
        int k = t & 31, n = t >> 5; // contiguous along K of B-rows
        async_load_f32((uint32_t)(uintptr_t)&Bs[k][n], Bm,
                       (uint32_t)(((size_t)(nBase + n) * ldb + kb + k) * 4));
      }
    } else if (MODE == 1) {
      #pragma unroll
      for (int t = tid; t < 64 * 32; t += 256) {
        int n = t & 63, k = t >> 6;
        async_load_f32((uint32_t)(uintptr_t)&Bs[k][n], Bm,
                       (uint32_t)(((size_t)(kb + k) * ldb + nBase + n) * 4));
      }
    } else {
      // guarded (zero-padded) tiles: regular path
      #pragma unroll
      for (int t = tid; t < 64 * 32; t += 256) {
        int n = t & 63, k = t >> 6;
        int col = nBase + n;
        int krow = kb + k;
        float v;
        if (MODE == 2) v = (col < Ntrue) ? Bm[(size_t)krow * ldb + col] : 0.0f;
        else           v = (krow < Kreal) ? Bm[(size_t)krow * ldb + col] : 0.0f;
        Bs[k][n] = v;
      }
    }
    async_wait0();
    __syncthreads();

    #pragma unroll
    for (int kk = 0; kk < 32; kk += 4) {
      const int ka = kk + lh * 2;
      v2f a, b0, b1;
      a.x = As[m0 + l15][ka];       a.y = As[m0 + l15][ka + 1];
      b0.x = Bs[ka][n0 + l15];      b0.y = Bs[ka + 1][n0 + l15];
      b1.x = Bs[ka][n0 + 16 + l15]; b1.y = Bs[ka + 1][n0 + 16 + l15];
      acc0 = wmma4(a, b0, acc0);
      acc1 = wmma4(a, b1, acc1);
    }
    __syncthreads();
  }

  // epilogue: C/D layout -> vgpr r holds row m = m0 + lh*8 + r
  #pragma unroll
  for (int r = 0; r < 8; ++r) {
    const int row = rowBase + m0 + lh * 8 + r;
    const int col0 = nBase + n0 + l15;
    const int col1 = col0 + 16;
    float v0 = acc0[r];
    float v1 = acc1[r];
    if (MODE == 0) {
      float cm = 1.0f - (float)mask[row];
      C[(size_t)row * ldc + col0] = v0 * cm;
      C[(size_t)row * ldc + col1] = v1 * cm;
    } else if (MODE == 1) {
      C[(size_t)row * ldc + col0] = v0;
      C[(size_t)row * ldc + col1] = v1;
    } else if (MODE == 2) {
      C[(size_t)row * ldc + col0] = (col0 < Ntrue) ? fmaxf(v0 + bias[col0], 0.0f) : 0.0f;
      C[(size_t)row * ldc + col1] = (col1 < Ntrue) ? fmaxf(v1 + bias[col1], 0.0f) : 0.0f;
    } else {
      C[(size_t)row * ldc + col0] = v0 + bias[col0] + resid[(size_t)row * ldc + col0];
      C[(size_t)row * ldc + col1] = v1 + bias[col1] + resid[(size_t)row * ldc + col1];
    }
  }
}

// ---------------- softmax + entropy per row (P=2048) ----------------
__launch_bounds__(256)
__global__ void softmax_entropy_kernel(float* __restrict__ probs,
                                       const int* __restrict__ mask,
                                       float* __restrict__ entropy) {
  __shared__ float red[256];
  const int tid = threadIdx.x;
  const int row = blockIdx.x;
  float* base = probs + (size_t)row * PNUM;
  float x[8];
  float m = -3.4e38f;
  #pragma unroll
  for (int j = 0; j < 8; ++j) { x[j] = base[tid + j * 256]; m = fmaxf(m, x[j]); }
  red[tid] = m; __syncthreads();
  for (int s = 128; s > 0; s >>= 1) {
    if (tid < s) red[tid] = fmaxf(red[tid], red[tid + s]);
    __syncthreads();
  }
  m = red[0]; __syncthreads();
  float sum = 0.f;
  #pragma unroll
  for (int j = 0; j < 8; ++j) { x[j] = __expf(x[j] - m); sum += x[j]; }
  sum = block_reduce_sum(sum, red, tid);
  const float inv = 1.0f / sum;
  float ent = 0.f;
  #pragma unroll
  for (int j = 0; j < 8; ++j) {
    float p = x[j] * inv;
    base[tid + j * 256] = p;
    ent += p * __logf(p);
  }
  ent = block_reduce_sum(ent, red, tid);
  if (tid == 0) {
    float cm = 1.0f - (float)mask[row];
    entropy[row] = -cm * ent;
  }
}

// ---------------- LayerNorm over D=512 ----------------
__launch_bounds__(256)
__global__ void layernorm_kernel(const float* __restrict__ x,
                                 const float* __restrict__ addend,
                                 float* __restrict__ out,
                                 const float* __restrict__ g,
                                 const float* __restrict__ b) {
  __shared__ float red[256];
  const int tid = threadIdx.x;
  const size_t off = (size_t)blockIdx.x * DDIM;
  float v0 = x[off + tid];
  float v1 = x[off + tid + 256];
  if (addend) { v0 += addend[off + tid]; v1 += addend[off + tid + 256]; }
  float mu = block_reduce_sum(v0 + v1, red, tid) * (1.0f / DDIM);
  float d0 = v0 - mu, d1 = v1 - mu;
  float var = block_reduce_sum(d0 * d0 + d1 * d1, red, tid) * (1.0f / DDIM);
  float rstd = rsqrtf(var + 1e-6f);
  out[off + tid]       = d0 * rstd * g[tid] + b[tid];
  out[off + tid + 256] = d1 * rstd * g[tid + 256] + b[tid + 256];
}

// ---------------- row squared-norms ----------------
__launch_bounds__(256)
__global__ void rowsq_kernel(const float* __restrict__ X, float* __restrict__ r, int rows) {
  int i = blockIdx.x * 256 + threadIdx.x;
  if (i >= rows) return;
  const float4* p4 = (const float4*)(X + (size_t)i * DDIM);
  float s = 0.f;
  for (int k = 0; k < DDIM / 4; ++k) {
    float4 v = p4[k];
    s += v.x * v.x + v.y * v.y + v.z * v.z + v.w * v.w;
  }
  r[i] = s;
}

// ---------------- d_pc via WMMA: 16 phenotype rows/block, streaming top-50 ----------------
// Block computes dot(phe[16 rows], emb^T) in 128-col tiles on the matrix pipe,
// converts to sqrt-distances in LDS, and 16 selector threads per row keep a
// streaming top-50 without materializing the 2048x40000 matrix.
__launch_bounds__(256)
__global__ void pc_dist_wmma_kernel(const float* __restrict__ phe,
                                    const float* __restrict__ emb,
                                    const float* __restrict__ rp,
                                    const float* __restrict__ rc,
                                    float* __restrict__ out_sum) {
  __shared__ float Ap[16][516];     // 16 phenotype rows x full K=512
  __shared__ float Bs[32][PC_NT + 4];
  __shared__ float Dt[16][PC_NT];
  __shared__ float rps[16];
  __shared__ float pool[256 * KC];
  __shared__ float redv[256];
  __shared__ int   redi[256];

  const int tid = threadIdx.x;
  const int lane = tid & 31;
  const int wave = tid >> 5;       // 8 waves: wave w owns cols [16w,16w+16)
  const int lh = lane >> 4;
  const int l15 = lane & 15;
  const int p0 = blockIdx.x * 16;

  // stage A (async) and rp
  for (int t = tid; t < 16 * DDIM; t += 256) {
    int r = t >> 9, c = t & 511;
    async_load_f32((uint32_t)(uintptr_t)&Ap[r][c], phe,
                   (uint32_t)(((size_t)(p0 + r) * DDIM + c) * 4));
  }
  if (tid < 16) rps[tid] = rp[p0 + tid];
  async_wait0();
  __syncthreads();

  const int selRow = tid & 15;   // selector: 16 threads per phenotype row
  const int selSlot = tid >> 4;
  float best[KC];
  #pragma unroll
  for (int j = 0; j < KC; ++j) best[j] = 3.0e38f;
  float worst = 3.0e38f;
  int wi = 0;

  for (int c0 = 0; c0 < VNUM; c0 += PC_NT) {
    v8f acc = {0.f, 0.f, 0.f, 0.f, 0.f, 0.f, 0.f, 0.f};
    const bool full = (c0 + PC_NT) <= VNUM;
    for (int kb = 0; kb < DDIM; kb += 32) {
      if (full) {
        #pragma unroll
        for (int t = tid; t < 32 * PC_NT; t += 256) {
          int k = t & 31, n = t >> 5;
          async_load_f32((uint32_t)(uintptr_t)&Bs[k][n], emb,
                         (uint32_t)(((size_t)(c0 + n) * DDIM + kb + k) * 4));
        }
      } else {
        #pragma unroll
        for (int t = tid; t < 32 * PC_NT; t += 256) {
          int k = t & 31, n = t >> 5;
          Bs[k][n] = (c0 + n < VNUM) ? emb[(size_t)(c0 + n) * DDIM + kb + k] : 0.0f;
        }
      }
      async_wait0();
      __syncthreads();
      #pragma unroll
      for (int kk = 0; kk < 32; kk += 4) {
        const int ka = kk + lh * 2;
        v2f a, b;
        a.x = Ap[l15][kb + ka];       a.y = Ap[l15][kb + ka + 1];
        b.x = Bs[ka][wave * 16 + l15]; b.y = Bs[ka + 1][wave * 16 + l15];
        acc = wmma4(a, b, acc);
      }
      __syncthreads();
    }
    // distances for this 16x128 tile
    {
      const int col = c0 + wave * 16 + l15;
      const bool valid = col < VNUM;
      const float rcv = valid ? rc[col] : 0.0f;
      #pragma unroll
      for (int r = 0; r < 8; ++r) {
        const int m = lh * 8 + r;
        Dt[m][wave * 16 + l15] =
            valid ? sqrtf(rps[m] + rcv - 2.0f * acc[r]) : 3.0e38f;
      }
    }
    __syncthreads();
    // streaming selection: each selector thread scans 8 columns of its row
    #pragma unroll
    for (int j = 0; j < 8; ++j) {
      float d = Dt[selRow][selSlot + j * 16];
      if (d < worst) {
        best[wi] = d;
        worst = -3.0e38f;
        for (int q = 0; q < KC; ++q)
          if (best[q] > worst) { worst = best[q]; wi = q; }
      }
    }
    // Dt is not rewritten until after the next K-loop's barriers -> safe
  }

  // merge 16 selector lists per row -> 50 smallest per row -> block sum
  for (int j = 0; j < KC; ++j) pool[tid * KC + j] = best[j];
  __syncthreads();
  float rowsum = 0.f;
  for (int it = 0; it < KC; ++it) {
    float lmin = 3.0e38f; int li = tid * KC;
    for (int j = 0; j < KC; ++j) {
      float v = pool[tid * KC + j];
      if (v < lmin) { lmin = v; li = tid * KC + j; }
    }
    redv[tid] = lmin; redi[tid] = li;
    __syncthreads();
    if (selSlot == 0) {                 // one thread per row
      float gmin = 3.0e38f; int gi = tid * KC;
      for (int s = 0; s < 16; ++s) {
        float v = redv[selRow + s * 16];
        if (v < gmin) { gmin = v; gi = redi[selRow + s * 16]; }
      }
      rowsum += gmin;
      pool[gi] = 3.0e38f;
    }
    __syncthreads();
  }
  if (selSlot == 0) redv[selRow] = rowsum;
  __syncthreads();
  if (tid == 0) {
    float s = 0.f;
    for (int m = 0; m < 16; ++m) s += redv[m];
    out_sum[blockIdx.x] = s;
  }
}

// ---------------- d_pp: per phenotype row, stats of 10 smallest sq-distances ----------------
__launch_bounds__(256)
__global__ void pp_dist_kernel(const float* __restrict__ phe,
                               const float* __restrict__ rp,
                               float* __restrict__ s_sum,
                               float* __restrict__ s_exp,
                               float* __restrict__ s_sq) {
  __shared__ __align__(16) float prow[DDIM];
  __shared__ float pool[256 * KP];
  __shared__ float redv[256];
  __shared__ int   redi[256];
  const int tid = threadIdx.x;
  const int p = blockIdx.x;
  for (int i = tid; i < DDIM; i += 256) prow[i] = phe[(size_t)p * DDIM + i];
  __syncthreads();
  const float rpp = rp[p];
  const float4* pr4 = (const float4*)prow;

  float best[KP];
  #pragma unroll
  for (int j = 0; j < KP; ++j) best[j] = 3.0e38f;
  float worst = 3.0e38f;
  int wi = 0;

  for (int q = tid; q < PNUM; q += 256) {
    const float4* e4 = (const float4*)(phe + (size_t)q * DDIM);
    float acc = 0.f;
    for (int k = 0; k < DDIM / 4; ++k) {
      float4 a = pr4[k]; float4 e = e4[k];
      acc += a.x * e.x + a.y * e.y + a.z * e.z + a.w * e.w;
    }
    float d2 = rpp + rp[q] - 2.0f * acc; // squared distance (includes self ~0)
    if (d2 < worst) {
      best[wi] = d2;
      worst = -3.0e38f;
      for (int j = 0; j < KP; ++j)
        if (best[j] > worst) { worst = best[j]; wi = j; }
    }
  }
  for (int j = 0; j < KP; ++j) pool[tid * KP + j] = best[j];
  __syncthreads();

  float sum = 0.f, sexp = 0.f, ssq = 0.f;
  for (int it = 0; it < KP; ++it) {
    float lmin = 3.0e38f; int li = tid * KP;
    for (int j = 0; j < KP; ++j) {
      float vv = pool[tid * KP + j];
      if (vv < lmin) { lmin = vv; li = tid * KP + j; }
    }
    redv[tid] = lmin; redi[tid] = li; __syncthreads();
    for (int s = 128; s > 0; s >>= 1) {
      if (tid < s && redv[tid + s] < redv[tid]) { redv[tid] = redv[tid + s]; redi[tid] = redi[tid + s]; }
      __syncthreads();
    }
    if (tid == 0) {
      float m = redv[0];
      sum += m; sexp += __expf(-m); ssq += m * m;
      pool[redi[0]] = 3.0e38f;
    }
    __syncthreads();
  }
  if (tid == 0) { s_sum[p] = sum; s_exp[p] = sexp; s_sq[p] = ssq; }
}

// ---------------- deterministic final reduction of scalars ----------------
__launch_bounds__(256)
__global__ void finalize_kernel(const float* __restrict__ pc,
                                const float* __restrict__ pps,
                                const float* __restrict__ ppe,
                                const float* __restrict__ ppq,
                                float* __restrict__ out) {
  __shared__ float red[256];
  const int tid = threadIdx.x;
  float a = 0.f, b = 0.f, c = 0.f, e = 0.f;
  if (tid < PC_BLOCKS) a = pc[tid];
  for (int j = tid; j < PNUM; j += 256) { b += pps[j]; c += ppe[j]; e += ppq[j]; }
  float sa = block_reduce_sum(a, red, tid);
  float sb = block_reduce_sum(b, red, tid);
  float sc = block_reduce_sum(c, red, tid);
  float se = block_reduce_sum(e, red, tid);
  if (tid == 0) {
    out[OUT_CDIST] = sa / (float)(PNUM * KC);
    float mean = sb / (float)(PNUM * KP);
    out[OUT_MET] = mean;
    out[OUT_INV] = sc / (float)(PNUM * KP);
    out[OUT_VAR] = se / (float)(PNUM * KP) - mean * mean;
  }
}

extern "C" void kernel_launch(void* const* d_in, const int* in_sizes, int n_in,
                              void* d_out_v, int out_size, void* d_ws, size_t ws_size,
                              hipStream_t stream) {
  const float* visit = (const float*)d_in[0];
  const int*   mask  = (const int*)d_in[1];
  const float* emb   = (const float*)d_in[2];
  const float* phe   = (const float*)d_in[3];
  const float* w1    = (const float*)d_in[4];
  const float* b1    = (const float*)d_in[5];
  const float* w2    = (const float*)d_in[6];
  const float* b2    = (const float*)d_in[7];
  const float* ln1g  = (const float*)d_in[8];
  const float* ln1b  = (const float*)d_in[9];
  const float* ln2g  = (const float*)d_in[10];
  const float* ln2b  = (const float*)d_in[11];

  float* out = (float*)d_out_v;
  float* ws  = (float*)d_ws;

  float* out2    = out + OUT_OUT2;
  float* probs   = out + OUT_PROBS;
  float* entropy = out + OUT_ENT;

  float* ctx = ws + WS_CTX;  // ctx, then out1 in place
  float* h   = ws + WS_H;
  float* rc  = ws + WS_RC;
  float* rp  = ws + WS_RP;
  float* pcS = ws + WS_PC;
  float* ppS = ws + WS_PPS;
  float* ppE = ws + WS_PPE;
  float* ppQ = ws + WS_PPQ;

  dim3 blk(256);

  // 1) masked logits -> probs buffer (pre-softmax)  [8192 x 2048, K=512]
  gemm_kernel<0><<<dim3(BT / 64, PNUM / 64), blk, 0, stream>>>(
      visit, phe, probs, mask, nullptr, nullptr,
      DDIM, DDIM, PNUM, DDIM, DDIM, PNUM);

  // 2) softmax over P + entropy
  softmax_entropy_kernel<<<dim3(BT), blk, 0, stream>>>(probs, mask, entropy);

  // 3) ctx = probs @ phe  [8192 x 512, K=2048]
  gemm_kernel<1><<<dim3(BT / 64, DDIM / 64), blk, 0, stream>>>(
      probs, phe, ctx, nullptr, nullptr, nullptr,
      PNUM, PNUM, DDIM, PNUM, DDIM, DDIM);

  // 4) out1 = LN1(visit + ctx), in place
  layernorm_kernel<<<dim3(BT), blk, 0, stream>>>(ctx, visit, ctx, ln1g, ln1b);

  // 5) h = relu(out1 @ w1 + b1)  [8192 x 2148 padded to 2176, K=512]
  gemm_kernel<2><<<dim3(BT / 64, DFF_PAD / 64), blk, 0, stream>>>(
      ctx, w1, h, nullptr, b1, nullptr,
      DDIM, DDIM, DFF, DDIM, DFF, DFF_PAD);

  // 6) pre2 = h @ w2 + b2 + out1 -> out2 region  [8192 x 512, K=2176(2148 real)]
  gemm_kernel<3><<<dim3(BT / 64, DDIM / 64), blk, 0, stream>>>(
      h, w2, out2, nullptr, b2, ctx,
      DFF_PAD, DFF, DDIM, DFF_PAD, DDIM, DDIM);

  // 7) out2 = LN2(pre2), in place
  layernorm_kernel<<<dim3(BT), blk, 0, stream>>>(out2, nullptr, out2, ln2g, ln2b);

  // 8) row squared norms
  rowsq_kernel<<<dim3((VNUM + 255) / 256), blk, 0, stream>>>(emb, rc, VNUM);
  rowsq_kernel<<<dim3((PNUM + 255) / 256), blk, 0, stream>>>(phe, rp, PNUM);

  // 9) d_pc on the matrix pipe, streaming top-50
  pc_dist_wmma_kernel<<<dim3(PC_BLOCKS), blk, 0, stream>>>(phe, emb, rp, rc, pcS);

  // 10) d_pp streaming top-10 (squared distances)
  pp_dist_kernel<<<dim3(PNUM), blk, 0, stream>>>(phe, rp, ppS, ppE, ppQ);

  // 11) deterministic scalar reduction
  finalize_kernel<<<dim3(1), blk, 0, stream>>>(pcS, ppS, ppE, ppQ, out);
}